// MegNetGraphConv_52209622450458
// MI455X (gfx1250) — compile-verified
//
#include <hip/hip_runtime.h>

// MegNet graph conv for gfx1250 (MI455X), fp32 WMMA path, M=32 per wave.
// d_in order: edge_feat, node_feat, graph_attr, src, dst,
//             eW1, eb1, eW2, eb2, nW1, nb1, nW2, nb2, aW1, ab1, aW2, ab2
// d_out: e_new [E,64] ++ v_new [N,64] ++ u_new [1,64]

typedef __attribute__((ext_vector_type(2))) float v2f;
typedef __attribute__((ext_vector_type(8))) float v8f;

#define WAVE_M    32    // rows per wave (two 16-row WMMA sub-tiles)
#define WAVES_PB  2
#define IN_PITCH  132   // 128 + 4 pad (kills 64-bank conflicts on column reads)
#define HID_PITCH 68    // 64 + 4 pad
#define HDIM      64

__device__ __forceinline__ float softplus_f(float x) {
    // softplus(x) = max(x,0) + log(1 + exp(-|x|))
    // Raw v_exp_f32 / v_log_f32 (base-2) with explicit scaling; the log argument
    // lies in (1,2], so no denorm/range guards are needed.
#if __has_builtin(__builtin_amdgcn_exp2f) && __has_builtin(__builtin_amdgcn_logf)
    float t = __builtin_amdgcn_exp2f(-fabsf(x) * 1.442695040888963f); // exp(-|x|)
    return fmaxf(x, 0.0f) +
           0.6931471805599453f * __builtin_amdgcn_logf(1.0f + t);    // ln(1+t)
#else
    return fmaxf(x, 0.0f) + __logf(1.0f + __expf(-fabsf(x)));
#endif
}

// 32x16 output tile: two 16x16 WMMA accumulators sharing one B fragment.
// A from LDS (rows l15 and l15+16), B from global weights (L2/WGP$-resident).
// Accumulators are pre-loaded with the per-column bias.
__device__ __forceinline__ void wmma_gemm2(const float* __restrict__ smat, int pitch,
                                           int ksteps, const float* __restrict__ W,
                                           int ntile, int lane, float bias,
                                           v8f& c0, v8f& c1) {
    const int half = lane >> 4;        // 0: lanes 0-15, 1: lanes 16-31
    const int l15  = lane & 15;
    const float* arow0 = smat + l15 * pitch;
    const float* arow1 = smat + (l15 + 16) * pitch;
    const float* wcol  = W + ntile * 16 + l15;
#pragma unroll
    for (int j = 0; j < 8; ++j) { c0[j] = bias; c1[j] = bias; }
#pragma unroll 4
    for (int ks = 0; ks < ksteps; ++ks) {
        const int k0 = 4 * ks + 2 * half;      // this lane's K rows: k0, k0+1
        v2f b;  b.x  = wcol[(k0)     * HDIM];
                b.y  = wcol[(k0 + 1) * HDIM];
        v2f a0; a0.x = arow0[k0];  a0.y = arow0[k0 + 1];
        v2f a1; a1.x = arow1[k0];  a1.y = arow1[k0 + 1];
        c0 = __builtin_amdgcn_wmma_f32_16x16x4_f32(false, a0, false, b,
                                                   (short)0, c0, false, false);
        c1 = __builtin_amdgcn_wmma_f32_16x16x4_f32(false, a1, false, b,
                                                   (short)0, c1, false, false);
    }
}

__global__ __launch_bounds__(WAVES_PB * 32)
void edge_mlp_kernel(const float* __restrict__ edge_feat,
                     const float* __restrict__ node_feat,
                     const float* __restrict__ gattr,
                     const int*   __restrict__ src,
                     const int*   __restrict__ dst,
                     const float* __restrict__ W1, const float* __restrict__ b1,
                     const float* __restrict__ W2, const float* __restrict__ b2,
                     float* __restrict__ e_out,   // [E,64]
                     float* __restrict__ esum,    // [N,64] accumulator (pre-zeroed)
                     float* __restrict__ cnt,     // [N]    accumulator (pre-zeroed)
                     int E)
{
    __shared__ float s_in [WAVES_PB][WAVE_M * IN_PITCH];
    __shared__ float s_hid[WAVES_PB][WAVE_M * HID_PITCH];

    const int  lane = threadIdx.x & 31;
    const int  wave = threadIdx.x >> 5;
    const long e0   = ((long)blockIdx.x * WAVES_PB + wave) * WAVE_M;
    if (e0 >= (long)E) return;                 // wave-uniform

    float* sin_ = s_in[wave];
    float* shid = s_hid[wave];

    // ---- stage input tile [32][128] = vi | vj | e | u ----
#pragma unroll 4
    for (int i = 0; i < WAVE_M; ++i) {
        long e  = e0 + i;
        long ec = (e < (long)E) ? e : (long)(E - 1);
        int  s  = src[ec];                     // wave-uniform -> scalar load
        int  d  = dst[ec];
        sin_[i * IN_PITCH + lane]      = node_feat[(long)s * 32 + lane];
        sin_[i * IN_PITCH + 32 + lane] = node_feat[(long)d * 32 + lane];
        sin_[i * IN_PITCH + 64 + lane] = edge_feat[ec * 32 + lane];
        sin_[i * IN_PITCH + 96 + lane] = gattr[lane];
    }

    const int half = lane >> 4;
    const int l15  = lane & 15;

    // ---- layer 1: [32,128] x [128,64] ----
    for (int t = 0; t < 4; ++t) {
        const int col = t * 16 + l15;
        v8f c0, c1;
        wmma_gemm2(sin_, IN_PITCH, 32, W1, t, lane, b1[col], c0, c1);
#pragma unroll
        for (int j = 0; j < 8; ++j) {
            int row = j + 8 * half;
            shid[(row)      * HID_PITCH + col] = softplus_f(c0[j]);
            shid[(row + 16) * HID_PITCH + col] = softplus_f(c1[j]);
        }
    }

    // dst indices for this lane's 16 output rows
    int dnA[8], dnB[8];
#pragma unroll
    for (int j = 0; j < 8; ++j) {
        long eA = e0 + j + 8 * half;
        long eB = eA + 16;
        dnA[j] = (eA < (long)E) ? dst[eA] : 0;
        dnB[j] = (eB < (long)E) ? dst[eB] : 0;
    }

    // ---- layer 2: [32,64] x [64,64], fused scatter-add by dst ----
    for (int t = 0; t < 4; ++t) {
        const int col = t * 16 + l15;
        v8f c0, c1;
        wmma_gemm2(shid, HID_PITCH, 16, W2, t, lane, b2[col], c0, c1);
#pragma unroll
        for (int j = 0; j < 8; ++j) {
            int  row = j + 8 * half;
            long eA  = e0 + row;
            long eB  = eA + 16;
            if (eA < (long)E) {
                float v = softplus_f(c0[j]);
                e_out[eA * 64 + col] = v;
                atomicAdd(&esum[(long)dnA[j] * 64 + col], v);
            }
            if (eB < (long)E) {
                float v = softplus_f(c1[j]);
                e_out[eB * 64 + col] = v;
                atomicAdd(&esum[(long)dnB[j] * 64 + col], v);
            }
        }
    }

    {   // edge count per dst (all 32 lanes -> 32 edges of this tile)
        long e = e0 + lane;
        if (e < (long)E) atomicAdd(&cnt[dst[e]], 1.0f);
    }
}

__global__ __launch_bounds__(WAVES_PB * 32)
void node_mlp_kernel(const float* __restrict__ node_feat,
                     const float* __restrict__ gattr,
                     const float* __restrict__ esum,   // [N,64]
                     const float* __restrict__ cnt,    // [N]
                     const float* __restrict__ W1, const float* __restrict__ b1,
                     const float* __restrict__ W2, const float* __restrict__ b2,
                     float* __restrict__ v_out,   // [N,64]
                     float* __restrict__ ue_acc,  // [64]  sum_n esum[n] (== sum_e e_new)
                     float* __restrict__ vsum,    // [64]  sum_n v_new[n]
                     int N)
{
    __shared__ float s_in [WAVES_PB][WAVE_M * IN_PITCH];
    __shared__ float s_hid[WAVES_PB][WAVE_M * HID_PITCH];

    const int  lane = threadIdx.x & 31;
    const int  wave = threadIdx.x >> 5;
    const long n0   = ((long)blockIdx.x * WAVES_PB + wave) * WAVE_M;
    if (n0 >= (long)N) return;

    float* sin_ = s_in[wave];
    float* shid = s_hid[wave];

    // ---- stage [32][128] = v | esum/max(cnt,1) | u ; fold ue reduction ----
    float ue_lo = 0.0f, ue_hi = 0.0f;
#pragma unroll 4
    for (int i = 0; i < WAVE_M; ++i) {
        long n  = n0 + i;
        bool ok = (n < (long)N);
        long nc = ok ? n : 0;
        float cdiv = 1.0f / fmaxf(cnt[nc], 1.0f);
        float elo = esum[nc * 64 + lane];
        float ehi = esum[nc * 64 + 32 + lane];
        sin_[i * IN_PITCH + lane]      = ok ? node_feat[nc * 32 + lane] : 0.0f;
        sin_[i * IN_PITCH + 32 + lane] = ok ? elo * cdiv : 0.0f;
        sin_[i * IN_PITCH + 64 + lane] = ok ? ehi * cdiv : 0.0f;
        sin_[i * IN_PITCH + 96 + lane] = gattr[lane];
        if (ok) { ue_lo += elo; ue_hi += ehi; }
    }
    atomicAdd(&ue_acc[lane],      ue_lo);
    atomicAdd(&ue_acc[32 + lane], ue_hi);

    const int half = lane >> 4;
    const int l15  = lane & 15;

    // ---- layer 1 ----
    for (int t = 0; t < 4; ++t) {
        const int col = t * 16 + l15;
        v8f c0, c1;
        wmma_gemm2(sin_, IN_PITCH, 32, W1, t, lane, b1[col], c0, c1);
#pragma unroll
        for (int j = 0; j < 8; ++j) {
            int row = j + 8 * half;
            shid[(row)      * HID_PITCH + col] = softplus_f(c0[j]);
            shid[(row + 16) * HID_PITCH + col] = softplus_f(c1[j]);
        }
    }

    // ---- layer 2, fused column-sum for uv readout ----
    for (int t = 0; t < 4; ++t) {
        const int col = t * 16 + l15;
        v8f c0, c1;
        wmma_gemm2(shid, HID_PITCH, 16, W2, t, lane, b2[col], c0, c1);
#pragma unroll
        for (int j = 0; j < 8; ++j) {
            int  row = j + 8 * half;
            long nA  = n0 + row;
            long nB  = nA + 16;
            if (nA < (long)N) {
                float v = softplus_f(c0[j]);
                v_out[nA * 64 + col] = v;
                atomicAdd(&vsum[col], v);
            }
            if (nB < (long)N) {
                float v = softplus_f(c1[j]);
                v_out[nB * 64 + col] = v;
                atomicAdd(&vsum[col], v);
            }
        }
    }
}

__global__ void attr_mlp_kernel(const float* __restrict__ gattr,
                                const float* __restrict__ ue_acc,
                                const float* __restrict__ vsum,
                                const float* __restrict__ W1, const float* __restrict__ b1,
                                const float* __restrict__ W2, const float* __restrict__ b2,
                                float* __restrict__ u_out, float invE, float invN)
{
    __shared__ float a_in[160];
    __shared__ float h1[64];
    const int t = threadIdx.x;   // 64 threads
    if (t < 32) a_in[t] = gattr[t];
    a_in[32 + t] = ue_acc[t] * invE;
    a_in[96 + t] = vsum[t]   * invN;
    __syncthreads();
    float s = b1[t];
    for (int k = 0; k < 160; ++k) s += a_in[k] * W1[k * 64 + t];
    h1[t] = softplus_f(s);
    __syncthreads();
    float s2 = b2[t];
    for (int k = 0; k < 64; ++k) s2 += h1[k] * W2[k * 64 + t];
    u_out[t] = softplus_f(s2);
}

__global__ void zero_kernel(float* __restrict__ p, long n) {
    long i = (long)blockIdx.x * blockDim.x + threadIdx.x;
    if (i < n) p[i] = 0.0f;
}

extern "C" void kernel_launch(void* const* d_in, const int* in_sizes, int n_in,
                              void* d_out, int out_size, void* d_ws, size_t ws_size,
                              hipStream_t stream) {
    const float* edge_feat = (const float*)d_in[0];
    const float* node_feat = (const float*)d_in[1];
    const float* gattr     = (const float*)d_in[2];
    const int*   src       = (const int*)  d_in[3];
    const int*   dst       = (const int*)  d_in[4];
    const float* eW1 = (const float*)d_in[5];  const float* eb1 = (const float*)d_in[6];
    const float* eW2 = (const float*)d_in[7];  const float* eb2 = (const float*)d_in[8];
    const float* nW1 = (const float*)d_in[9];  const float* nb1 = (const float*)d_in[10];
    const float* nW2 = (const float*)d_in[11]; const float* nb2 = (const float*)d_in[12];
    const float* aW1 = (const float*)d_in[13]; const float* ab1 = (const float*)d_in[14];
    const float* aW2 = (const float*)d_in[15]; const float* ab2 = (const float*)d_in[16];

    const int E = in_sizes[0] / 32;
    const int N = in_sizes[1] / 32;

    float* out_e = (float*)d_out;                       // [E,64]
    float* out_v = out_e + (long)E * 64;                // [N,64]
    float* out_u = out_v + (long)N * 64;                // [64]

    float* esum = (float*)d_ws;                         // [N,64]
    float* cnt  = esum + (long)N * 64;                  // [N]
    float* ue   = cnt + N;                              // [64]
    float* vs   = ue + 64;                              // [64]

    const long ztotal = (long)N * 64 + N + 128;
    zero_kernel<<<(int)((ztotal + 255) / 256), 256, 0, stream>>>(esum, ztotal);

    const long etiles = ((long)E + WAVE_M - 1) / WAVE_M;
    const int  eblk   = (int)((etiles + WAVES_PB - 1) / WAVES_PB);
    edge_mlp_kernel<<<eblk, WAVES_PB * 32, 0, stream>>>(
        edge_feat, node_feat, gattr, src, dst,
        eW1, eb1, eW2, eb2, out_e, esum, cnt, E);

    const long ntiles = ((long)N + WAVE_M - 1) / WAVE_M;
    const int  nblk   = (int)((ntiles + WAVES_PB - 1) / WAVES_PB);
    node_mlp_kernel<<<nblk, WAVES_PB * 32, 0, stream>>>(
        node_feat, gattr, esum, cnt,
        nW1, nb1, nW2, nb2, out_v, ue, vs, N);

    attr_mlp_kernel<<<1, 64, 0, stream>>>(
        gattr, ue, vs, aW1, ab1, aW2, ab2, out_u,
        1.0f / (float)E, 1.0f / (float)N);
}